// TwoTowerGNN_24799141167260
// MI455X (gfx1250) — compile-verified
//
#include <hip/hip_runtime.h>
#include <hip/hip_bf16.h>
#include <math.h>

// Problem constants (match reference)
constexpr int Bn  = 64;          // batch
constexpr int Nn  = 200;         // nodes per graph
constexpr int NT  = Bn * Nn;     // 12800 flattened nodes
constexpr int En  = 409600;      // edges
constexpr int SD  = 1024;        // sent dim
constexpr int CID = 256;         // concept in dim
constexpr int CD  = 200;         // concept dim
constexpr int Hh  = 4;           // heads
constexpr int DH  = 50;          // dim per head

typedef __attribute__((ext_vector_type(2))) float v2f;
typedef __attribute__((ext_vector_type(8))) float v8f;

__device__ __forceinline__ float gelu_tanh(float x) {
  // jax.nn.gelu default (approximate=True)
  float x3 = x * x * x;
  return 0.5f * x * (1.0f + tanhf(0.7978845608028654f * (x + 0.044715f * x3)));
}

// ---------------------------------------------------------------------------
// Generic fp32 WMMA GEMM: C[M,N] = act(A[M,K] @ B[K,N] + bias)
// One wave per 16x32 output slab (two 16x16 WMMA tiles sharing one A fragment),
// K-loop in steps of 4 via V_WMMA_F32_16X16X4_F32.
// Optional row gather (A row m = table[gatherIds[m]*lda ...]) for the embedding
// tower. M must be a multiple of 16 and K a multiple of 4 (true at all sites).
// Edge columns are handled branch-free: clamp address into bounds, zero the
// value with a select (v_cndmask) -> no per-iteration EXEC save/restore.
// Per ISA 7.12.2 (f32 layouts):
//   A 16x4 : lanes 0-15 rows, VGPR0/1 = K0/K1; lanes 16-31 = K2/K3   -> v2f
//   B 4x16 : lanes 0-15 cols, VGPR0/1 = K0/K1; lanes 16-31 = K2/K3   -> v2f
//   C 16x16: VGPR v: lanes 0-15 -> row v, lanes 16-31 -> row v+8     -> v8f
// ---------------------------------------------------------------------------
__global__ void k_gemm_wmma(const float* __restrict__ A, const float* __restrict__ Bm,
                            float* __restrict__ C, const float* __restrict__ bias,
                            const int* __restrict__ gatherIds, const float* __restrict__ table,
                            int M, int N, int K, int lda, int ldb, int ldc, int act) {
  const int wave = (int)((blockIdx.x * blockDim.x + threadIdx.x) >> 5);
  const int lane = threadIdx.x & 31;
  const int tilesN  = (N + 15) >> 4;
  const int tilesNP = (tilesN + 1) >> 1;        // pairs of N-tiles
  const int tilesM  = M >> 4;
  if (wave >= tilesM * tilesNP) return;         // wave-uniform: EXEC stays all-ones
  const int tm = wave / tilesNP;
  const int tp = wave % tilesNP;
  const int hl = lane >> 4;                     // 0 -> K pair {0,1}; 1 -> {2,3}
  const int l  = lane & 15;

  const int rowA = tm * 16 + l;
  const float* Ar = gatherIds ? (table + (size_t)gatherIds[rowA] * (size_t)lda)
                              : (A + (size_t)rowA * (size_t)lda);

  const int col0 = tp * 32 + l;
  const int col1 = col0 + 16;
  const bool ok0 = (col0 < N);
  const bool ok1 = (col1 < N);
  const int c0 = ok0 ? col0 : (N - 1);          // clamped, always in-bounds
  const int c1 = ok1 ? col1 : (N - 1);

  v8f acc0 = {};
  v8f acc1 = {};
  for (int kk = 0; kk < K; kk += 4) {
    const int ka = kk + (hl << 1);
    v2f a; a.x = Ar[ka]; a.y = Ar[ka + 1];
    const float* Brow = Bm + (size_t)ka * (size_t)ldb;
    float b0x = Brow[c0], b0y = Brow[(size_t)ldb + c0];
    float b1x = Brow[c1], b1y = Brow[(size_t)ldb + c1];
    v2f b0; b0.x = ok0 ? b0x : 0.0f; b0.y = ok0 ? b0y : 0.0f;
    v2f b1; b1.x = ok1 ? b1x : 0.0f; b1.y = ok1 ? b1y : 0.0f;
    acc0 = __builtin_amdgcn_wmma_f32_16x16x4_f32(false, a, false, b0,
                                                 (short)0, acc0, false, false);
    acc1 = __builtin_amdgcn_wmma_f32_16x16x4_f32(false, a, false, b1,
                                                 (short)0, acc1, false, false);
  }

  const int rbase = tm * 16 + hl * 8;
  if (ok0) {
    const float bb = bias ? bias[col0] : 0.0f;
    for (int v = 0; v < 8; ++v) {
      float val = acc0[v] + bb;
      if (act) val = gelu_tanh(val);
      C[(size_t)(rbase + v) * (size_t)ldc + col0] = val;
    }
  }
  if (ok1) {
    const float bb = bias ? bias[col1] : 0.0f;
    for (int v = 0; v < 8; ++v) {
      float val = acc1[v] + bb;
      if (act) val = gelu_tanh(val);
      C[(size_t)(rbase + v) * (size_t)ldc + col1] = val;
    }
  }
}

// ---------------------------------------------------------------------------
// CSR-by-destination build (shared by both GAT layers)
// ---------------------------------------------------------------------------
__global__ void k_zero_i32(int* __restrict__ p, int n) {
  int i = blockIdx.x * blockDim.x + threadIdx.x;
  if (i < n) p[i] = 0;
}

__global__ void k_count(const int* __restrict__ dst, int* __restrict__ counts) {
  int e = blockIdx.x * blockDim.x + threadIdx.x;
  if (e < En) atomicAdd(&counts[dst[e]], 1);
}

__global__ void k_scan(const int* __restrict__ counts, int* __restrict__ row_ptr,
                       int* __restrict__ cursor, int n) {
  __shared__ int buf[1024];
  __shared__ int carry;
  const int t = threadIdx.x;
  if (t == 0) carry = 0;
  __syncthreads();
  for (int base = 0; base < n; base += 1024) {
    int i = base + t;
    int v = (i < n) ? counts[i] : 0;
    buf[t] = v;
    __syncthreads();
    for (int off = 1; off < 1024; off <<= 1) {
      int add = (t >= off) ? buf[t - off] : 0;
      __syncthreads();
      buf[t] += add;
      __syncthreads();
    }
    int incl = buf[t];
    int c = carry;
    if (i < n) { row_ptr[i] = c + incl - v; cursor[i] = c + incl - v; }
    __syncthreads();
    if (t == 1023) carry = c + buf[1023];
    __syncthreads();
  }
  if (t == 0) row_ptr[n] = carry;
}

__global__ void k_fill(const int* __restrict__ dst, int* __restrict__ cursor,
                       int* __restrict__ esort) {
  int e = blockIdx.x * blockDim.x + threadIdx.x;
  if (e < En) {
    int pos = atomicAdd(&cursor[dst[e]], 1);
    esort[pos] = e;
  }
}

// ---------------------------------------------------------------------------
// Per-(node,head) attention pre-scores: s = <h[n,head,:], a[head,:]>
// ---------------------------------------------------------------------------
__global__ void k_scores(const float* __restrict__ h, const float* __restrict__ a_src,
                         const float* __restrict__ a_dst, float* __restrict__ ssrc,
                         float* __restrict__ sdst) {
  int idx = blockIdx.x * blockDim.x + threadIdx.x;
  if (idx >= NT * Hh) return;
  int n = idx >> 2, hh = idx & 3;
  const float* hp = h + (size_t)n * CD + hh * DH;
  const float* as = a_src + hh * DH;
  const float* ad = a_dst + hh * DH;
  float s1 = 0.0f, s2 = 0.0f;
  for (int d = 0; d < DH; ++d) { float v = hp[d]; s1 += v * as[d]; s2 += v * ad[d]; }
  ssrc[idx] = s1;
  sdst[idx] = s2;
}

// ---------------------------------------------------------------------------
// GAT softmax + aggregation, one workgroup per destination node.
// Two passes over the node's incoming edge list (CSR): max, then exp/accumulate.
// out[n,:] = (sum_e ex_e * h[src_e,:]) / (den + 1e-16), 0 for deg==0 nodes.
// No float atomics; all gathers hit L2 (10 MB feature array << 192 MB L2).
// ---------------------------------------------------------------------------
__global__ void k_gat_agg(const float* __restrict__ h, const float* __restrict__ ssrc,
                          const float* __restrict__ sdst, const float* __restrict__ e_emb,
                          const int* __restrict__ etype, const int* __restrict__ srcIds,
                          const int* __restrict__ row_ptr, const int* __restrict__ esort,
                          float* __restrict__ out, int applyGelu) {
  const int n = blockIdx.x;
  const int t = threadIdx.x;
  const int start = row_ptr[n];
  const int deg = row_ptr[n + 1] - start;

  __shared__ float red[256];
  __shared__ float mhead[Hh];
  __shared__ float denh[Hh];
  __shared__ float exs[64 * Hh];
  __shared__ int   srcs[64];

  float sd[Hh];
  for (int k = 0; k < Hh; ++k) sd[k] = sdst[n * Hh + k];

  // Pass 1: per-head max of leaky_relu(s_src + s_dst + e_emb)
  float mloc[Hh];
  for (int k = 0; k < Hh; ++k) mloc[k] = -INFINITY;
  for (int i = t; i < deg; i += 256) {
    int e = esort[start + i];
    int s = srcIds[e];
    const float* eb = e_emb + etype[e] * Hh;
    for (int k = 0; k < Hh; ++k) {
      float val = ssrc[s * Hh + k] + sd[k] + eb[k];
      val = (val > 0.0f) ? val : 0.2f * val;
      mloc[k] = fmaxf(mloc[k], val);
    }
  }
  for (int k = 0; k < Hh; ++k) {
    red[t] = mloc[k];
    __syncthreads();
    for (int off = 128; off > 0; off >>= 1) {
      if (t < off) red[t] = fmaxf(red[t], red[t + off]);
      __syncthreads();
    }
    if (t == 0) mhead[k] = (deg > 0) ? red[0] : 0.0f;   // reference: nonfinite max -> 0
    __syncthreads();
  }

  // Pass 2: chunked exp + feature accumulation (thread t owns feature dim t)
  float acc = 0.0f;
  float denloc[Hh] = {0.0f, 0.0f, 0.0f, 0.0f};
  const int hd = (t < CD) ? (t / DH) : 0;
  for (int base = 0; base < deg; base += 64) {
    int len = min(64, deg - base);
    if (t < len) {
      int e = esort[start + base + t];
      int s = srcIds[e];
      srcs[t] = s;
      __builtin_prefetch((const void*)(h + (size_t)s * CD), 0, 0);  // global_prefetch_b8
      const float* eb = e_emb + etype[e] * Hh;
      for (int k = 0; k < Hh; ++k) {
        float val = ssrc[s * Hh + k] + sd[k] + eb[k];
        val = (val > 0.0f) ? val : 0.2f * val;
        float ex = expf(val - mhead[k]);
        exs[t * Hh + k] = ex;
        denloc[k] += ex;
      }
    }
    __syncthreads();
    if (t < CD) {
      for (int j = 0; j < len; ++j)
        acc += exs[j * Hh + hd] * h[(size_t)srcs[j] * CD + t];
    }
    __syncthreads();
  }
  for (int k = 0; k < Hh; ++k) {
    red[t] = denloc[k];
    __syncthreads();
    for (int off = 128; off > 0; off >>= 1) {
      if (t < off) red[t] += red[t + off];
      __syncthreads();
    }
    if (t == 0) denh[k] = red[0] + 1e-16f;
    __syncthreads();
  }
  if (t < CD) {
    float v = (deg > 0) ? acc / denh[hd] : 0.0f;
    if (applyGelu) v = gelu_tanh(v);              // fused inter-layer GELU
    out[(size_t)n * CD + t] = v;
  }
}

// ---------------------------------------------------------------------------
// Cosine similarity: one wave per (b, n); out = (cos + 1) / 2
// ---------------------------------------------------------------------------
__global__ void k_cos(const float* __restrict__ sentp, const float* __restrict__ node,
                      float* __restrict__ out) {
  const int wid = (int)((blockIdx.x * blockDim.x + threadIdx.x) >> 5);
  const int lane = threadIdx.x & 31;
  if (wid >= NT) return;
  const int b = wid / Nn;
  const float* sp = sentp + (size_t)b * CD;
  const float* nd = node + (size_t)wid * CD;
  float num = 0.0f, na = 0.0f, nb = 0.0f;
  for (int f = lane; f < CD; f += 32) {
    float s = sp[f], v = nd[f];
    num += s * v; na += s * s; nb += v * v;
  }
  for (int off = 16; off > 0; off >>= 1) {
    num += __shfl_xor(num, off, 32);
    na  += __shfl_xor(na,  off, 32);
    nb  += __shfl_xor(nb,  off, 32);
  }
  if (lane == 0) {
    float denom = fmaxf(sqrtf(na), 1e-8f) * fmaxf(sqrtf(nb), 1e-8f);
    out[wid] = (num / denom + 1.0f) * 0.5f;
  }
}

// ---------------------------------------------------------------------------
extern "C" void kernel_launch(void* const* d_in, const int* in_sizes, int n_in,
                              void* d_out, int out_size, void* d_ws, size_t ws_size,
                              hipStream_t stream) {
  (void)in_sizes; (void)n_in; (void)out_size; (void)ws_size;
  const float* sent_vecs   = (const float*)d_in[0];
  const int*   concept_ids = (const int*)d_in[1];
  // d_in[2] node_type_ids, d_in[3] node_scores, d_in[4] adj_lengths: unused by reference
  const int*   edge_idx    = (const int*)d_in[5];
  const int*   etype       = (const int*)d_in[6];
  const float* emb_table   = (const float*)d_in[7];
  const float* cpt_W       = (const float*)d_in[8];
  const float* cpt_b       = (const float*)d_in[9];
  const float* sent_W      = (const float*)d_in[10];
  const float* sent_b      = (const float*)d_in[11];
  const float* gat_W1      = (const float*)d_in[12];
  const float* a_src1      = (const float*)d_in[13];
  const float* a_dst1      = (const float*)d_in[14];
  const float* e_emb1      = (const float*)d_in[15];
  const float* gat_W2      = (const float*)d_in[16];
  const float* a_src2      = (const float*)d_in[17];
  const float* a_dst2      = (const float*)d_in[18];
  const float* e_emb2      = (const float*)d_in[19];
  const int* srcIds = edge_idx;        // edge_index_ids[0]
  const int* dstIds = edge_idx + En;   // edge_index_ids[1]

  // Workspace carve-out (~32.5 MB total)
  char* w = (char*)d_ws;
  auto take = [&](size_t bytes) -> void* {
    void* p = (void*)w;
    w += (bytes + 255) & ~(size_t)255;
    return p;
  };
  float* x_buf  = (float*)take(sizeof(float) * (size_t)NT * CD);
  float* h_buf  = (float*)take(sizeof(float) * (size_t)NT * CD);
  float* o_buf  = (float*)take(sizeof(float) * (size_t)NT * CD);
  float* sent_p = (float*)take(sizeof(float) * (size_t)Bn * CD);
  float* ssrc   = (float*)take(sizeof(float) * (size_t)NT * Hh);
  float* sdst   = (float*)take(sizeof(float) * (size_t)NT * Hh);
  int* counts   = (int*)take(sizeof(int) * NT);
  int* row_ptr  = (int*)take(sizeof(int) * (NT + 1));
  int* cursor   = (int*)take(sizeof(int) * NT);
  int* esort    = (int*)take(sizeof(int) * En);

  const int TPB = 256;
  const int tilesNP = (((CD + 15) / 16) + 1) / 2;   // 7 pairs of N-tiles

  // --- CSR build by destination (shared by both GAT layers) ---
  k_zero_i32<<<(NT + TPB - 1) / TPB, TPB, 0, stream>>>(counts, NT);
  k_count<<<(En + TPB - 1) / TPB, TPB, 0, stream>>>(dstIds, counts);
  k_scan<<<1, 1024, 0, stream>>>(counts, row_ptr, cursor, NT);
  k_fill<<<(En + TPB - 1) / TPB, TPB, 0, stream>>>(dstIds, cursor, esort);

  // --- Sentence tower: sent_p = gelu(sent_vecs @ sent_W + b) ---
  {
    int waves = (Bn / 16) * tilesNP;
    k_gemm_wmma<<<(waves * 32 + TPB - 1) / TPB, TPB, 0, stream>>>(
        sent_vecs, sent_W, sent_p, sent_b, nullptr, nullptr,
        Bn, CD, SD, SD, CD, CD, /*act=*/1);
  }
  // --- Concept tower: x = gelu(emb_table[ids] @ cpt_W + b) ---
  {
    int waves = (NT / 16) * tilesNP;
    k_gemm_wmma<<<(waves * 32 + TPB - 1) / TPB, TPB, 0, stream>>>(
        nullptr, cpt_W, x_buf, cpt_b, concept_ids, emb_table,
        NT, CD, CID, CID, CD, CD, /*act=*/1);
  }

  const int gatWaves = (NT / 16) * tilesNP;
  // --- GAT layer 1 (output gets fused GELU = input of layer 2) ---
  k_gemm_wmma<<<(gatWaves * 32 + TPB - 1) / TPB, TPB, 0, stream>>>(
      x_buf, gat_W1, h_buf, nullptr, nullptr, nullptr,
      NT, CD, CD, CD, CD, CD, /*act=*/0);
  k_scores<<<(NT * Hh + TPB - 1) / TPB, TPB, 0, stream>>>(h_buf, a_src1, a_dst1, ssrc, sdst);
  k_gat_agg<<<NT, TPB, 0, stream>>>(h_buf, ssrc, sdst, e_emb1, etype, srcIds,
                                    row_ptr, esort, o_buf, /*applyGelu=*/1);

  // --- GAT layer 2 ---
  k_gemm_wmma<<<(gatWaves * 32 + TPB - 1) / TPB, TPB, 0, stream>>>(
      o_buf, gat_W2, h_buf, nullptr, nullptr, nullptr,
      NT, CD, CD, CD, CD, CD, /*act=*/0);
  k_scores<<<(NT * Hh + TPB - 1) / TPB, TPB, 0, stream>>>(h_buf, a_src2, a_dst2, ssrc, sdst);
  k_gat_agg<<<NT, TPB, 0, stream>>>(h_buf, ssrc, sdst, e_emb2, etype, srcIds,
                                    row_ptr, esort, x_buf, /*applyGelu=*/0);

  // --- Cosine head ---
  k_cos<<<(NT * 32 + TPB - 1) / TPB, TPB, 0, stream>>>(sent_p, x_buf, (float*)d_out);
}